// LabelPropagation_45122926412032
// MI455X (gfx1250) — compile-verified
//
#include <hip/hip_runtime.h>
#include <hip/hip_bf16.h>
#include <math.h>
#include <stdint.h>

#define N_TOT   2560
#define D_FEATC 8192
#define N_CLS   128
#define N_SUP   640
#define N_QRY   1920
#define K_TOP   20
#define CG_ITERS 160
#define EPS_F   2.220446049250313e-16f

typedef float v2f __attribute__((ext_vector_type(2)));
typedef float v8f __attribute__((ext_vector_type(8)));

// D = A(16x4,f32) * B(4x16,f32) + C(16x16,f32) — CDNA5 V_WMMA_F32_16X16X4_F32
__device__ __forceinline__ v8f wmma4(v2f a, v2f b, v8f c) {
    return __builtin_amdgcn_wmma_f32_16x16x4_f32(
        false, a, false, b, (short)0, c, false, false);
}

// CDNA5 async DMA: global -> LDS, 16 bytes per lane, tracked by ASYNCcnt.
__device__ __forceinline__ void async_load_b128(const float* gptr, unsigned lds_addr) {
    asm volatile("global_load_async_to_lds_b128 %0, %1, off"
                 :: "v"(lds_addr), "v"((unsigned long long)(uintptr_t)gptr)
                 : "memory");
}
__device__ __forceinline__ void wait_async0() {
    asm volatile("s_wait_asynccnt 0x0" ::: "memory");
}

// ---------------- row squared norms ----------------
__global__ __launch_bounds__(256) void rownorm_k(const float* __restrict__ feats,
                                                 float* __restrict__ sq) {
    __shared__ float red[256];
    int row = blockIdx.x;
    const float* f = feats + (size_t)row * D_FEATC;
    float s = 0.f;
    for (int j = threadIdx.x; j < D_FEATC; j += 256) { float v = f[j]; s += v * v; }
    red[threadIdx.x] = s; __syncthreads();
    for (int off = 128; off > 0; off >>= 1) {
        if (threadIdx.x < off) red[threadIdx.x] += red[threadIdx.x + off];
        __syncthreads();
    }
    if (threadIdx.x == 0) sq[row] = red[0];
}

// ---------------- Gram + affinity W = exp(-dist/2), 64x64 tile per block ----------------
// Double-buffered async global->LDS staging of 64x32 panels (16B-aligned rows: 36 floats).
__global__ __launch_bounds__(128) void gramW_k(const float* __restrict__ feats,
                                               const float* __restrict__ sq,
                                               float* __restrict__ W) {
    __shared__ float As[2][64][36];
    __shared__ float Bs[2][64][36];
    const int ti = blockIdx.y * 64;
    const int tj = blockIdx.x * 64;
    const int t = threadIdx.x;
    const int lane = t & 31;
    const int wave = t >> 5;
    const int m  = lane & 15;     // A-row / B-col / C-col index
    const int kh = lane >> 4;     // k-half select

    auto stage = [&](int buf, int kk) {
        #pragma unroll
        for (int u = 0; u < 4; u++) {
            int id = t + 128 * u;       // 0..511 : 64 rows x 8 chunks of 16B
            int r  = id >> 3;
            int c4 = (id & 7) * 4;
            async_load_b128(feats + (size_t)(ti + r) * D_FEATC + kk + c4,
                            (unsigned)(uintptr_t)&As[buf][r][c4]);
            async_load_b128(feats + (size_t)(tj + r) * D_FEATC + kk + c4,
                            (unsigned)(uintptr_t)&Bs[buf][r][c4]);
        }
    };

    stage(0, 0);
    v8f acc[4] = {};
    for (int kk = 0; kk < D_FEATC; kk += 32) {
        int buf = (kk >> 5) & 1;
        wait_async0();        // this wave's async loads for `buf` complete
        __syncthreads();      // all waves' loads done; all prior compute done
        if (kk + 32 < D_FEATC) stage(buf ^ 1, kk + 32);
        #pragma unroll
        for (int k2 = 0; k2 < 32; k2 += 4) {
            v2f a;
            a.x = As[buf][wave*16 + m][k2 + 2*kh];
            a.y = As[buf][wave*16 + m][k2 + 2*kh + 1];
            #pragma unroll
            for (int jb = 0; jb < 4; jb++) {
                v2f b;
                b.x = Bs[buf][jb*16 + m][k2 + 2*kh];
                b.y = Bs[buf][jb*16 + m][k2 + 2*kh + 1];
                acc[jb] = wmma4(a, b, acc[jb]);
            }
        }
    }
    const float inv_d = 1.0f / (float)D_FEATC;
    int i0 = ti + wave*16;
    #pragma unroll
    for (int jb = 0; jb < 4; jb++) {
        int j = tj + jb*16 + m;
        float sqj = sq[j];
        #pragma unroll
        for (int r = 0; r < 8; r++) {
            int i = i0 + r + 8*kh;
            float g = acc[jb][r];
            float dist = (sq[i] + sqj - 2.0f * g) * inv_d;
            W[(size_t)i * N_TOT + j] = expf(-0.5f * dist);
        }
    }
}

// ---------------- per-row top-K -> mask (lowest-index tie break, like lax.top_k) ----------------
__global__ __launch_bounds__(256) void topk_k(const float* __restrict__ W,
                                              unsigned char* __restrict__ mask) {
    __shared__ float vals[N_TOT];
    __shared__ float redv[256];
    __shared__ int   redi[256];
    int row = blockIdx.x;
    const float* wr = W + (size_t)row * N_TOT;
    unsigned char* mr = mask + (size_t)row * N_TOT;
    for (int j = threadIdx.x; j < N_TOT; j += 256) { vals[j] = wr[j]; mr[j] = 0; }
    __syncthreads();
    for (int it = 0; it < K_TOP; it++) {
        float bv = -INFINITY; int bi = 0x7fffffff;
        for (int j = threadIdx.x; j < N_TOT; j += 256) {
            float v = vals[j];
            if (v > bv || (v == bv && j < bi)) { bv = v; bi = j; }
        }
        redv[threadIdx.x] = bv; redi[threadIdx.x] = bi;
        __syncthreads();
        for (int off = 128; off > 0; off >>= 1) {
            if (threadIdx.x < off) {
                float ov = redv[threadIdx.x + off]; int oi = redi[threadIdx.x + off];
                if (ov > redv[threadIdx.x] ||
                    (ov == redv[threadIdx.x] && oi < redi[threadIdx.x])) {
                    redv[threadIdx.x] = ov; redi[threadIdx.x] = oi;
                }
            }
            __syncthreads();
        }
        if (threadIdx.x == 0) { mr[redi[0]] = 1; vals[redi[0]] = -INFINITY; }
        __syncthreads();
    }
}

// ---------------- symmetric mask apply (in-place) + degree ----------------
__global__ __launch_bounds__(256) void symD_k(float* __restrict__ W,
                                              const unsigned char* __restrict__ mask,
                                              float* __restrict__ Dv) {
    __shared__ float red[256];
    int i = blockIdx.x;
    float s = 0.f;
    for (int j = threadIdx.x; j < N_TOT; j += 256) {
        int mm = mask[(size_t)i * N_TOT + j] | mask[(size_t)j * N_TOT + i];
        float w = mm ? W[(size_t)i * N_TOT + j] : 0.f;
        W[(size_t)i * N_TOT + j] = w;
        s += w;
    }
    red[threadIdx.x] = s; __syncthreads();
    for (int off = 128; off > 0; off >>= 1) {
        if (threadIdx.x < off) red[threadIdx.x] += red[threadIdx.x + off];
        __syncthreads();
    }
    if (threadIdx.x == 0) Dv[i] = red[0];
}

__global__ void dsi_k(const float* __restrict__ Dv, float* __restrict__ dsi) {
    int i = blockIdx.x * blockDim.x + threadIdx.x;
    if (i < N_TOT) dsi[i] = sqrtf(1.0f / (Dv[i] + EPS_F));
}

// ---------------- A = I - alpha * S + EPS (overwrites W in place) ----------------
__global__ __launch_bounds__(256) void buildA_k(float* __restrict__ W,
                                                const float* __restrict__ dsi,
                                                const float* __restrict__ alpha) {
    size_t idx = (size_t)blockIdx.x * 256 + threadIdx.x;
    if (idx >= (size_t)N_TOT * N_TOT) return;
    int i = (int)(idx / N_TOT), j = (int)(idx % N_TOT);
    float s = dsi[i] * W[idx] * dsi[j];
    W[idx] = ((i == j) ? 1.0f : 0.0f) - alpha[0] * s + EPS_F;
}

// ---------------- CG setup ----------------
__global__ void zero_scalars_k(float* s, int n) {
    int t = blockIdx.x * blockDim.x + threadIdx.x;
    if (t < n) s[t] = 0.f;
}

__global__ __launch_bounds__(256) void cginit_k(const int* __restrict__ slab,
                                                float* X, float* R, float* P, float* rr) {
    int idx = blockIdx.x * 256 + threadIdx.x;
    if (idx >= N_TOT * N_CLS) return;
    int i = idx >> 7, c = idx & 127;
    float y = (i < N_SUP && slab[i] == c) ? 1.0f : 0.0f;
    X[idx] = 0.f; R[idx] = y; P[idx] = y;
    if (y != 0.f) atomicAdd(&rr[c], 1.0f);
}

// ---------------- AP = A @ P  (2560x2560 @ 2560x128) via f32 WMMA ----------------
// P panel (32x128) staged with async global->LDS (row stride 132 floats = 33 x 16B).
__global__ __launch_bounds__(128) void gemmAP_k(const float* __restrict__ A,
                                                const float* __restrict__ P,
                                                float* __restrict__ AP) {
    __shared__ float Ps[32][132];
    int bi = blockIdx.x * 64;
    int lane = threadIdx.x & 31, wave = threadIdx.x >> 5;
    int m = lane & 15, kh = lane >> 4;
    v8f acc[8] = {};
    const float* arow = A + (size_t)(bi + wave*16 + m) * N_TOT;
    for (int kk = 0; kk < N_TOT; kk += 32) {
        __syncthreads();   // previous compute done before overwriting Ps
        #pragma unroll
        for (int u = 0; u < 8; u++) {
            int id = threadIdx.x + 128 * u;   // 0..1023 : 32 rows x 32 chunks of 16B
            int r  = id >> 5;
            int c4 = (id & 31) * 4;
            async_load_b128(P + (size_t)(kk + r) * N_CLS + c4,
                            (unsigned)(uintptr_t)&Ps[r][c4]);
        }
        wait_async0();
        __syncthreads();
        #pragma unroll
        for (int k2 = 0; k2 < 32; k2 += 4) {
            v2f a;
            a.x = arow[kk + k2 + 2*kh];
            a.y = arow[kk + k2 + 2*kh + 1];
            #pragma unroll
            for (int cb = 0; cb < 8; cb++) {
                v2f b;
                b.x = Ps[k2 + 2*kh][cb*16 + m];
                b.y = Ps[k2 + 2*kh + 1][cb*16 + m];
                acc[cb] = wmma4(a, b, acc[cb]);
            }
        }
    }
    int i0 = bi + wave*16;
    #pragma unroll
    for (int cb = 0; cb < 8; cb++)
        #pragma unroll
        for (int r = 0; r < 8; r++)
            AP[(size_t)(i0 + r + 8*kh) * N_CLS + cb*16 + m] = acc[cb][r];
}

// ---------------- per-column dot: out[c] += sum_i U[i,c]*V[i,c] ----------------
__global__ __launch_bounds__(256) void coldot_k(const float* __restrict__ U,
                                                const float* __restrict__ V,
                                                float* __restrict__ out) {
    __shared__ float red[256];
    int c = threadIdx.x & 127, half = threadIdx.x >> 7;
    int r0 = blockIdx.x * 32;
    float s = 0.f;
    for (int r = r0 + half; r < r0 + 32; r += 2) {
        size_t idx = (size_t)r * N_CLS + c;
        s += U[idx] * V[idx];
    }
    red[threadIdx.x] = s; __syncthreads();
    if (threadIdx.x < 128) atomicAdd(&out[c], red[threadIdx.x] + red[threadIdx.x + 128]);
}

__global__ void alpha_k(const float* rr, const float* pAp, float* alphas, float* rrnew) {
    int c = threadIdx.x; if (c >= N_CLS) return;
    float d = pAp[c];
    alphas[c] = (d != 0.f) ? rr[c] / d : 0.f;
    rrnew[c] = 0.f;
}

__global__ __launch_bounds__(256) void xrupd_k(float* X, float* R,
                                               const float* __restrict__ P,
                                               const float* __restrict__ AP,
                                               const float* __restrict__ alphas,
                                               float* rrnew) {
    __shared__ float red[256];
    int c = threadIdx.x & 127, half = threadIdx.x >> 7;
    int r0 = blockIdx.x * 32;
    float a = alphas[c];
    float s = 0.f;
    for (int r = r0 + half; r < r0 + 32; r += 2) {
        size_t idx = (size_t)r * N_CLS + c;
        X[idx] += a * P[idx];
        float rv = R[idx] - a * AP[idx];
        R[idx] = rv;
        s += rv * rv;
    }
    red[threadIdx.x] = s; __syncthreads();
    if (threadIdx.x < 128) atomicAdd(&rrnew[c], red[threadIdx.x] + red[threadIdx.x + 128]);
}

__global__ void beta_k(float* rr, const float* rrnew, float* betas, float* pAp) {
    int c = threadIdx.x; if (c >= N_CLS) return;
    float o = rr[c];
    betas[c] = (o != 0.f) ? rrnew[c] / o : 0.f;
    rr[c] = rrnew[c];
    pAp[c] = 0.f;
}

__global__ __launch_bounds__(256) void pupd_k(float* P, const float* __restrict__ R,
                                              const float* __restrict__ betas) {
    int idx = blockIdx.x * 256 + threadIdx.x;
    if (idx >= N_TOT * N_CLS) return;
    int c = idx & 127;
    P[idx] = R[idx] + betas[c] * P[idx];
}

// ---------------- loss + accuracy ----------------
__global__ void loss_k(const float* __restrict__ X, const int* __restrict__ slab,
                       const int* __restrict__ qlab, float* acc2) {
    int r = blockIdx.x * blockDim.x + threadIdx.x;
    if (r >= N_TOT) return;
    const float* xr = X + (size_t)r * N_CLS;
    float mx = xr[0]; int am = 0;
    for (int c = 1; c < N_CLS; c++) { float v = xr[c]; if (v > mx) { mx = v; am = c; } }
    float se = 0.f;
    for (int c = 0; c < N_CLS; c++) se += expf(xr[c] - mx);
    float lse = mx + logf(se);
    int gt = (r < N_SUP) ? slab[r] : qlab[r - N_SUP];
    atomicAdd(&acc2[0], -(xr[gt] - lse));
    if (r >= N_SUP && am == qlab[r - N_SUP]) atomicAdd(&acc2[1], 1.0f);
}

__global__ void fin_k(const float* acc2, float* out) {
    if (threadIdx.x == 0) {
        out[0] = acc2[0] / (float)N_TOT;
        out[1] = acc2[1] / (float)N_QRY;
    }
}

extern "C" void kernel_launch(void* const* d_in, const int* in_sizes, int n_in,
                              void* d_out, int out_size, void* d_ws, size_t ws_size,
                              hipStream_t stream) {
    (void)in_sizes; (void)n_in; (void)out_size; (void)ws_size;
    const float* feats = (const float*)d_in[0];
    const float* alpha = (const float*)d_in[1];
    const int*   slab  = (const int*)d_in[2];
    const int*   qlab  = (const int*)d_in[3];
    // d_in[4] = k (==20), compiled in as K_TOP

    char* ws = (char*)d_ws;
    size_t off = 0;
    auto alloc = [&](size_t bytes) -> char* {
        char* p = ws + off;
        off += (bytes + 255) & ~(size_t)255;
        return p;
    };
    float*         sq    = (float*)alloc((size_t)N_TOT * 4);
    float*         W     = (float*)alloc((size_t)N_TOT * N_TOT * 4);   // later holds A
    unsigned char* mask  = (unsigned char*)alloc((size_t)N_TOT * N_TOT);
    float*         Dv    = (float*)alloc((size_t)N_TOT * 4);
    float*         dsi   = (float*)alloc((size_t)N_TOT * 4);
    float*         X     = (float*)alloc((size_t)N_TOT * N_CLS * 4);
    float*         R     = (float*)alloc((size_t)N_TOT * N_CLS * 4);
    float*         P     = (float*)alloc((size_t)N_TOT * N_CLS * 4);
    float*         AP    = (float*)alloc((size_t)N_TOT * N_CLS * 4);
    float*         scal  = (float*)alloc((5 * N_CLS + 2) * 4);
    float *rr = scal, *pAp = scal + 128, *alphas = scal + 256,
          *betas = scal + 384, *rrnew = scal + 512, *acc2 = scal + 640;

    rownorm_k<<<N_TOT, 256, 0, stream>>>(feats, sq);
    dim3 g1(N_TOT / 64, N_TOT / 64);
    gramW_k<<<g1, 128, 0, stream>>>(feats, sq, W);
    topk_k<<<N_TOT, 256, 0, stream>>>(W, mask);
    symD_k<<<N_TOT, 256, 0, stream>>>(W, mask, Dv);
    dsi_k<<<(N_TOT + 255) / 256, 256, 0, stream>>>(Dv, dsi);
    buildA_k<<<(int)(((size_t)N_TOT * N_TOT + 255) / 256), 256, 0, stream>>>(W, dsi, alpha);

    zero_scalars_k<<<1, 1024, 0, stream>>>(scal, 5 * N_CLS + 2);
    cginit_k<<<(N_TOT * N_CLS + 255) / 256, 256, 0, stream>>>(slab, X, R, P, rr);

    for (int it = 0; it < CG_ITERS; it++) {
        gemmAP_k<<<N_TOT / 64, 128, 0, stream>>>(W, P, AP);
        coldot_k<<<N_TOT / 32, 256, 0, stream>>>(P, AP, pAp);
        alpha_k<<<1, 128, 0, stream>>>(rr, pAp, alphas, rrnew);
        xrupd_k<<<N_TOT / 32, 256, 0, stream>>>(X, R, P, AP, alphas, rrnew);
        beta_k<<<1, 128, 0, stream>>>(rr, rrnew, betas, pAp);
        pupd_k<<<(N_TOT * N_CLS + 255) / 256, 256, 0, stream>>>(P, R, betas);
    }

    loss_k<<<(N_TOT + 255) / 256, 256, 0, stream>>>(X, slab, qlab, acc2);
    fin_k<<<1, 64, 0, stream>>>(acc2, (float*)d_out);
}